// GraphAttentionLayer_52218212385088
// MI455X (gfx1250) — compile-verified
//
#include <hip/hip_runtime.h>
#include <hip/hip_bf16.h>
#include <math.h>

typedef __attribute__((ext_vector_type(2))) float v2f;
typedef __attribute__((ext_vector_type(8))) float v8f;

#define N_NODES 8192
#define F_IN    128
#define F_OUT   64
#define NEG_INF_F (-9e15f)
#define ALPHA   0.2f

// ---------------------------------------------------------------------------
// K1: Wh = h @ W   (M=8192, K=128, N=64), f32 WMMA 16x16x4.
// Block = 128 threads = 4 waves; block covers 16 rows x 64 cols; wave w owns
// the 16-col tile starting at w*16.  A layout: lanes 0-15 K={0,1} in v[0],v[1],
// lanes 16-31 K={2,3}.  C/D: VGPR r holds M = r + 8*(lane>>4), N = lane&15.
// ---------------------------------------------------------------------------
__global__ void __launch_bounds__(128)
wh_gemm_kernel(const float* __restrict__ h, const float* __restrict__ W,
               float* __restrict__ Wh) {
  const int wave    = threadIdx.x >> 5;
  const int lane    = threadIdx.x & 31;
  const int rowBase = blockIdx.x * 16;
  const int colBase = wave * 16;
  const int row     = lane & 15;
  const int col     = lane & 15;
  const int kh      = (lane >> 4) << 1;   // 0 or 2

  v8f c = {};
  const float* hrow = h + (size_t)(rowBase + row) * F_IN;
  for (int k0 = 0; k0 < F_IN; k0 += 4) {
    v2f a = *(const v2f*)(hrow + k0 + kh);          // K=kh, kh+1 (8B aligned)
    v2f b;
    b.x = W[(size_t)(k0 + kh)     * F_OUT + colBase + col];
    b.y = W[(size_t)(k0 + kh + 1) * F_OUT + colBase + col];
    c = __builtin_amdgcn_wmma_f32_16x16x4_f32(false, a, false, b,
                                              (short)0, c, false, false);
  }
  const int mOff = (lane >> 4) * 8;
#pragma unroll
  for (int r = 0; r < 8; ++r)
    Wh[(size_t)(rowBase + mOff + r) * F_OUT + colBase + col] = c[r];
}

// ---------------------------------------------------------------------------
// K2: s1[i] = Wh[i,:] . a[0:64];  s2[i] = Wh[i,:] . a[64:128]
// ---------------------------------------------------------------------------
__global__ void __launch_bounds__(256)
scores_kernel(const float* __restrict__ Wh, const float* __restrict__ a,
              float* __restrict__ s1, float* __restrict__ s2) {
  const int i = blockIdx.x * blockDim.x + threadIdx.x;
  if (i >= N_NODES) return;
  float acc1 = 0.f, acc2 = 0.f;
  const float* w = Wh + (size_t)i * F_OUT;
#pragma unroll 8
  for (int f = 0; f < F_OUT; ++f) {
    acc1 += w[f] * a[f];
    acc2 += w[f] * a[F_OUT + f];
  }
  s1[i] = acc1;
  s2[i] = acc2;
}

// ---------------------------------------------------------------------------
// K3: one block (256 thr = 8 waves) per row i.
//   e_j = leakyrelu(s1_i + s2_j) if adj>0 else -9e15  -> staged in LDS (32KB)
//   row max, exp, row sum, write normalized attention.  adj read from HBM once.
// ---------------------------------------------------------------------------
__global__ void __launch_bounds__(256)
softmax_row_kernel(const int* __restrict__ adj, const float* __restrict__ s1,
                   const float* __restrict__ s2, float* __restrict__ att) {
  __shared__ float erow[N_NODES];   // 32 KB of the 320 KB WGP LDS
  __shared__ float red[8];

  const int i   = blockIdx.x;
  const int tid = threadIdx.x;
  const float s1i = s1[i];
  const long long base = (long long)i * N_NODES;

  // pass 1: masked scores into LDS + running max
  float lmax = NEG_INF_F;
  for (int j = tid; j < N_NODES; j += 256) {
    float v = NEG_INF_F;
    if (adj[base + j] > 0) {
      float t = s1i + s2[j];
      v = (t > 0.f) ? t : ALPHA * t;
    }
    erow[j] = v;
    lmax = fmaxf(lmax, v);
  }
#pragma unroll
  for (int o = 16; o > 0; o >>= 1) lmax = fmaxf(lmax, __shfl_xor(lmax, o, 32));
  if ((tid & 31) == 0) red[tid >> 5] = lmax;
  __syncthreads();
  if (tid == 0) {
    float m = red[0];
    for (int w = 1; w < 8; ++w) m = fmaxf(m, red[w]);
    red[0] = m;
  }
  __syncthreads();
  const float m = red[0];

  // pass 2: exp + sum (masked entries: exp(-9e15 - m) == 0)
  float lsum = 0.f;
  for (int j = tid; j < N_NODES; j += 256) {
    float p = __expf(erow[j] - m);
    erow[j] = p;
    lsum += p;
  }
#pragma unroll
  for (int o = 16; o > 0; o >>= 1) lsum += __shfl_xor(lsum, o, 32);
  __syncthreads();                       // red[] reuse hazard
  if ((tid & 31) == 0) red[tid >> 5] = lsum;
  __syncthreads();
  if (tid == 0) {
    float s = 0.f;
    for (int w = 0; w < 8; ++w) s += red[w];
    red[0] = 1.0f / s;
  }
  __syncthreads();
  const float inv = red[0];

  // pass 3: write normalized attention row
  for (int j = tid; j < N_NODES; j += 256)
    att[base + j] = erow[j] * inv;
}

// ---------------------------------------------------------------------------
// K4: h' = attention @ Wh   (M=8192, K=8192, N=64), f32 WMMA 16x16x4.
// Row tiles are processed in REVERSE order: K3 finished by writing the
// highest-index attention rows, so they are the ones still resident in the
// 192 MB L2.  Reading them first converts a large fraction of the 256 MB
// attention re-read into L2 hits instead of HBM misses.
// ---------------------------------------------------------------------------
__global__ void __launch_bounds__(128)
hprime_gemm_kernel(const float* __restrict__ att, const float* __restrict__ Wh,
                   float* __restrict__ hp) {
  const int wave    = threadIdx.x >> 5;
  const int lane    = threadIdx.x & 31;
  const int rowBase = (gridDim.x - 1 - blockIdx.x) * 16;   // reversed mapping
  const int colBase = wave * 16;
  const int row     = lane & 15;
  const int col     = lane & 15;
  const int kh      = (lane >> 4) << 1;

  v8f c = {};
  const float* arow = att + (long long)(rowBase + row) * N_NODES;
  for (int k0 = 0; k0 < N_NODES; k0 += 4) {
    v2f a = *(const v2f*)(arow + k0 + kh);           // contiguous pair, 8B aligned
    v2f b;
    b.x = Wh[(size_t)(k0 + kh)     * F_OUT + colBase + col];
    b.y = Wh[(size_t)(k0 + kh + 1) * F_OUT + colBase + col];
    c = __builtin_amdgcn_wmma_f32_16x16x4_f32(false, a, false, b,
                                              (short)0, c, false, false);
  }
  const int mOff = (lane >> 4) * 8;
#pragma unroll
  for (int r = 0; r < 8; ++r)
    hp[(size_t)(rowBase + mOff + r) * F_OUT + colBase + col] = c[r];
}

// ---------------------------------------------------------------------------
extern "C" void kernel_launch(void* const* d_in, const int* in_sizes, int n_in,
                              void* d_out, int out_size, void* d_ws, size_t ws_size,
                              hipStream_t stream) {
  const float* h   = (const float*)d_in[0];  // [8192,128]
  const int*   adj = (const int*)  d_in[1];  // [8192,8192]
  const float* W   = (const float*)d_in[2];  // [128,64]
  const float* a   = (const float*)d_in[3];  // [128,1]

  float* hp  = (float*)d_out;                       // h_prime [8192,64]
  float* att = hp + (size_t)N_NODES * F_OUT;        // attention [8192,8192]

  float* Wh = (float*)d_ws;                         // [8192,64] = 2 MB
  float* s1 = Wh + (size_t)N_NODES * F_OUT;         // [8192]
  float* s2 = s1 + N_NODES;                         // [8192]

  wh_gemm_kernel   <<<N_NODES / 16, 128, 0, stream>>>(h, W, Wh);
  scores_kernel    <<<N_NODES / 256, 256, 0, stream>>>(Wh, a, s1, s2);
  softmax_row_kernel<<<N_NODES,      256, 0, stream>>>(adj, s1, s2, att);
  hprime_gemm_kernel<<<N_NODES / 16, 128, 0, stream>>>(att, Wh, hp);
}